// ParallelEuclideanEmbed_38397007626567
// MI455X (gfx1250) — compile-verified
//
#include <hip/hip_runtime.h>
#include <math.h>

typedef float v2f __attribute__((ext_vector_type(2)));
typedef float v8f __attribute__((ext_vector_type(8)));

namespace {
constexpr int Sc = 4, Lc = 3;
constexpr int Dc = 256, DIc = 512;
constexpr int Nc = 16, Kc = 4, Rc = 16;
constexpr int INc = 32, Ec = 128;
constexpr int Bc = 8, Tc = 512;
constexpr int Mc = Bc * Tc;                      // 4096 rows (B*T)
constexpr int NT = 4;                            // 16-col n-tiles per wave
constexpr size_t HSZ  = (size_t)Mc * Dc;         // per-stream h        (4 MB)
constexpr size_t XZSZ = (size_t)Mc * 2 * DIc;    // per-stream xz       (16 MB)
constexpr size_t USZ  = (size_t)Mc * DIc;        // per-stream u        (8 MB)
constexpr size_t XDSZ = (size_t)Mc * 48;         // per-stream x_dbl
constexpr size_t HPSZ = (size_t)Bc * Dc;         // per-stream pooled h
}

// ---------------------------------------------------------------------------
// f32 WMMA GEMM, register-blocked: C[m,n] = sum_k A[m,k] * W[n,k] (+ bias[n])
// One wave computes a 16x64 strip of C (4 accumulators); each A fragment is
// reused by 4 V_WMMA_F32_16X16X4_F32 ops -> 4 WMMA per 5 global b64 loads.
// A-matrix VGPR layout (ISA 7.12.2, 32-bit A 16x4): lanes 0-15 hold K=0,1;
// lanes 16-31 hold K=2,3 (row = lane&15). B is the transposed arrangement,
// C/D: col = lane&15, row = vgpr + 8*(lane>>4).
// grid.z batches streams via element strides sA/sW/sC. Tile validity is
// wave-uniform, so EXEC is all-1s at every WMMA.
// ---------------------------------------------------------------------------
__global__ void wmma_gemm_f32(const float* __restrict__ A, const float* __restrict__ W,
                              const float* __restrict__ bias, float* __restrict__ C,
                              int M, int N, int Kd, int lda,
                              long sA, long sW, long sC) {
  A += (long)blockIdx.z * sA;
  W += (long)blockIdx.z * sW;
  C += (long)blockIdx.z * sC;
  const int lane = threadIdx.x & 31;
  const int wave = threadIdx.x >> 5;
  const int n0 = (blockIdx.x * 4 + wave) * (16 * NT);
  if (n0 >= N) return;
  const int m0 = blockIdx.y * 16;
  const int lc = lane & 15;     // A row / W row / C col within tile
  const int hi = lane >> 4;     // selects K pair {0,1} vs {2,3}
  const float* Arow = A + (size_t)(m0 + lc) * lda;

  const float* Wrow[NT];
  bool valid[NT];               // wave-uniform (n0, N uniform across the wave)
#pragma unroll
  for (int i = 0; i < NT; ++i) {
    const int n = n0 + i * 16;
    valid[i] = (n < N);                     // N is a multiple of 16
    Wrow[i] = W + (size_t)(valid[i] ? (n + lc) : lc) * Kd;
  }

  v8f acc[NT] = {};
  const int kiters = Kd >> 2;   // Kd is always a multiple of 4 here
  for (int k4 = 0; k4 < kiters; ++k4) {
    const int kk = (k4 << 2) + (hi << 1);
    const v2f a = *reinterpret_cast<const v2f*>(Arow + kk);
#pragma unroll
    for (int i = 0; i < NT; ++i) {
      const v2f b = *reinterpret_cast<const v2f*>(Wrow[i] + kk);
      acc[i] = __builtin_amdgcn_wmma_f32_16x16x4_f32(false, a, false, b, (short)0,
                                                     acc[i], false, false);
    }
  }
#pragma unroll
  for (int i = 0; i < NT; ++i) {
    if (!valid[i]) continue;
    const int nc = n0 + i * 16 + lc;
    const float bv = (bias != nullptr) ? bias[nc] : 0.0f;
#pragma unroll
    for (int r = 0; r < 8; ++r)
      C[(size_t)(m0 + r + hi * 8) * N + nc] = acc[i][r] + bv;
  }
}

// ---------------------------------------------------------------------------
// Depthwise causal conv (K=4) + bias + SiLU: reads xi = xz[:, :DIc] -> u
// ---------------------------------------------------------------------------
__global__ void conv_silu_kernel(const float* __restrict__ xz, const float* __restrict__ cw,
                                 const float* __restrict__ cb, float* __restrict__ u) {
  const int ls = blockIdx.z;
  const float* xzs = xz + (size_t)ls * XZSZ;
  float* us = u + (size_t)ls * USZ;
  const float* cws = cw + (size_t)ls * Lc * DIc * Kc;   // next stream, same layer
  const float* cbs = cb + (size_t)ls * Lc * DIc;
  const int idx = blockIdx.x * blockDim.x + threadIdx.x;
  if (idx >= Mc * DIc) return;
  const int d = idx & (DIc - 1);
  const int m = idx >> 9;            // DIc == 512
  const int tt = m & (Tc - 1);       // Tc == 512
  float acc = cbs[d];
#pragma unroll
  for (int k = 0; k < Kc; ++k) {
    const int ts = tt + k - (Kc - 1);
    if (ts >= 0)
      acc = fmaf(xzs[(size_t)(m + k - (Kc - 1)) * (2 * DIc) + d], cws[d * Kc + k], acc);
  }
  const float sg = 1.0f / (1.0f + expf(-acc));
  us[(size_t)m * DIc + d] = acc * sg;
}

// ---------------------------------------------------------------------------
// Fused selective scan: dt-proj + softplus, h_n = exp(dt*A_n)*h_n + dt*B_n*u,
// y_t = C . h + u*Dp, then y *= silu(z).  Writes y over xz[:, :DIc].
// One thread per (b, channel d); per-timestep x_dbl row (48 floats) staged in
// LDS and shared by all channels of the block.
// ---------------------------------------------------------------------------
__global__ void scan_kernel(float* __restrict__ xz, const float* __restrict__ u,
                            const float* __restrict__ xd,
                            const float* __restrict__ dtw, const float* __restrict__ dtb,
                            const float* __restrict__ alog, const float* __restrict__ dpar,
                            int l, int sg) {
  const int ls = blockIdx.z;
  const int s = sg + ls;
  const int b = blockIdx.y;
  const int d = blockIdx.x * blockDim.x + threadIdx.x;   // 0..DIc-1
  const size_t sl = (size_t)s * Lc + l;
  const float* wrow = dtw + (sl * DIc + d) * Rc;
  const float dtbv = dtb[sl * DIc + d];
  const float* arow = alog + (sl * DIc + d) * Nc;
  const float dpv = dpar[sl * DIc + d];
  float Av[Nc], wv[Rc], hs[Nc];
#pragma unroll
  for (int n = 0; n < Nc; ++n) { Av[n] = -expf(arow[n]); hs[n] = 0.0f; }
#pragma unroll
  for (int r = 0; r < Rc; ++r) wv[r] = wrow[r];

  const size_t rowbase = (size_t)ls * Mc + (size_t)b * Tc;
  const float* xdp = xd + rowbase * 48;
  const float* up = u + rowbase * DIc;
  float* xzp = xz + rowbase * (2 * DIc);

  __shared__ float sh[48];  // [0:16)=dt-rank inputs, [16:32)=B, [32:48)=C
  for (int t = 0; t < Tc; ++t) {
    if (threadIdx.x < 48) sh[threadIdx.x] = xdp[(size_t)t * 48 + threadIdx.x];
    __syncthreads();
    float dtr = dtbv;
#pragma unroll
    for (int r = 0; r < Rc; ++r) dtr = fmaf(wv[r], sh[r], dtr);
    const float dtv = (dtr > 20.0f) ? dtr : log1pf(expf(dtr));   // softplus
    const float uv = up[(size_t)t * DIc + d];
    const float du = dtv * uv;
    float y = 0.0f;
#pragma unroll
    for (int n = 0; n < Nc; ++n) {
      const float da = expf(dtv * Av[n]);
      hs[n] = fmaf(da, hs[n], du * sh[16 + n]);
      y = fmaf(hs[n], sh[32 + n], y);
    }
    y = fmaf(uv, dpv, y);
    const float z = xzp[(size_t)t * (2 * DIc) + DIc + d];
    const float zsig = z / (1.0f + expf(-z));                    // silu(z)
    xzp[(size_t)t * (2 * DIc) + d] = y * zsig;
    __syncthreads();
  }
}

// ---------------------------------------------------------------------------
// Mean pool over T, output head, and stream combine
// ---------------------------------------------------------------------------
__global__ void pool_kernel(const float* __restrict__ h, float* __restrict__ hp, int total) {
  const int idx = blockIdx.x * blockDim.x + threadIdx.x;
  if (idx >= total) return;
  const int dd = idx % Dc;
  const int b = (idx / Dc) % Bc;
  const int ls = idx / (Dc * Bc);
  const float* hs = h + (size_t)ls * HSZ + (size_t)b * Tc * Dc + dd;
  float acc = 0.0f;
  for (int t = 0; t < Tc; ++t) acc += hs[(size_t)t * Dc];
  hp[idx] = acc * (1.0f / (float)Tc);
}

__global__ void head_kernel(const float* __restrict__ hp, const float* __restrict__ opw,
                            const float* __restrict__ opb, float* __restrict__ out,
                            int sg, int total) {
  const int idx = blockIdx.x * blockDim.x + threadIdx.x;
  if (idx >= total) return;
  const int e = idx % Ec;
  const int b = (idx / Ec) % Bc;
  const int ls = idx / (Ec * Bc);
  const int s = sg + ls;
  const float* hrow = hp + (size_t)(ls * Bc + b) * Dc;
  const float* wrow = opw + ((size_t)s * Ec + e) * Dc;
  float acc = opb[s * Ec + e];
  for (int dd = 0; dd < Dc; ++dd) acc = fmaf(hrow[dd], wrow[dd], acc);
  out[((size_t)s * Bc + b) * Ec + e] = acc;
}

__global__ void combine_kernel(float* __restrict__ out) {
  const int idx = blockIdx.x * blockDim.x + threadIdx.x;
  if (idx >= Bc * Ec) return;
  float acc = 0.0f;
  for (int s = 0; s < Sc; ++s) acc += out[(size_t)s * Bc * Ec + idx];
  out[(size_t)Sc * Bc * Ec + idx] = acc;
}

// ---------------------------------------------------------------------------
extern "C" void kernel_launch(void* const* d_in, const int* in_sizes, int n_in,
                              void* d_out, int out_size, void* d_ws, size_t ws_size,
                              hipStream_t stream) {
  (void)in_sizes; (void)n_in; (void)out_size;
  const float* x_in[Sc] = {(const float*)d_in[0], (const float*)d_in[1],
                           (const float*)d_in[2], (const float*)d_in[3]};
  const float* in_w = (const float*)d_in[4];
  const float* cw   = (const float*)d_in[5];
  const float* cb   = (const float*)d_in[6];
  const float* xpw  = (const float*)d_in[7];
  const float* dtw  = (const float*)d_in[8];
  const float* dtb  = (const float*)d_in[9];
  const float* alog = (const float*)d_in[10];
  const float* dpar = (const float*)d_in[11];
  const float* ow   = (const float*)d_in[12];
  const float* ipw  = (const float*)d_in[13];
  const float* ipb  = (const float*)d_in[14];
  const float* opw  = (const float*)d_in[15];
  const float* opb  = (const float*)d_in[16];
  float* out = (float*)d_out;

  const size_t perS = (HSZ + XZSZ + USZ + XDSZ + HPSZ) * sizeof(float);
  const int ns = (ws_size >= (size_t)Sc * perS) ? Sc : 1;   // deterministic in ws_size

  float* hB  = (float*)d_ws;
  float* xzB = hB + (size_t)ns * HSZ;
  float* uB  = xzB + (size_t)ns * XZSZ;
  float* xdB = uB + (size_t)ns * USZ;
  float* hpB = xdB + (size_t)ns * XDSZ;

  const int colsPerBlock = 4 * 16 * NT;   // 4 waves x 64 columns
  for (int sg = 0; sg < Sc; sg += ns) {
    const int g = ns;
    // input projection: h = x @ ipw^T + ipb   (M=4096, N=256, K=32)
    for (int j = 0; j < g; ++j) {
      const int s = sg + j;
      wmma_gemm_f32<<<dim3((Dc + colsPerBlock - 1) / colsPerBlock, Mc / 16, 1),
                      128, 0, stream>>>(
          x_in[s], ipw + (size_t)s * Dc * INc, ipb + (size_t)s * Dc,
          hB + (size_t)j * HSZ, Mc, Dc, INc, INc, 0, 0, 0);
    }
    for (int l = 0; l < Lc; ++l) {
      const size_t sl0 = (size_t)(sg * Lc + l);
      // in_proj: xz = h @ in_w^T   (N=1024, K=256)
      wmma_gemm_f32<<<dim3((2 * DIc + colsPerBlock - 1) / colsPerBlock, Mc / 16, g),
                      128, 0, stream>>>(
          hB, in_w + sl0 * (2 * DIc) * Dc, nullptr, xzB,
          Mc, 2 * DIc, Dc, Dc,
          (long)HSZ, (long)Lc * (2 * DIc) * Dc, (long)XZSZ);
      // depthwise causal conv + SiLU -> u
      conv_silu_kernel<<<dim3((Mc * DIc) / 256, 1, g), 256, 0, stream>>>(
          xzB, cw + sl0 * DIc * Kc, cb + sl0 * DIc, uB);
      // x_proj: x_dbl = u @ xpw^T   (N=48, K=512)
      wmma_gemm_f32<<<dim3(1, Mc / 16, g), 128, 0, stream>>>(
          uB, xpw + sl0 * 48 * DIc, nullptr, xdB,
          Mc, 48, DIc, DIc,
          (long)USZ, (long)Lc * 48 * DIc, (long)XDSZ);
      // fused dt-proj + selective scan + gating, writes y into xz[:, :DIc]
      scan_kernel<<<dim3(DIc / 128, Bc, g), 128, 0, stream>>>(
          xzB, uB, xdB, dtw, dtb, alog, dpar, l, sg);
      // out_proj: h = y @ ow^T   (N=256, K=512, lda=1024 through xz buffer)
      wmma_gemm_f32<<<dim3((Dc + colsPerBlock - 1) / colsPerBlock, Mc / 16, g),
                      128, 0, stream>>>(
          xzB, ow + sl0 * Dc * DIc, nullptr, hB,
          Mc, Dc, DIc, 2 * DIc,
          (long)XZSZ, (long)Lc * Dc * DIc, (long)HSZ);
    }
    // temporal mean pool + output head (writes d_out rows [sg, sg+g))
    {
      const int totP = g * Bc * Dc;
      pool_kernel<<<(totP + 255) / 256, 256, 0, stream>>>(hB, hpB, totP);
      const int totH = g * Bc * Ec;
      head_kernel<<<(totH + 255) / 256, 256, 0, stream>>>(hpB, opw, opb, out, sg, totH);
    }
  }
  combine_kernel<<<(Bc * Ec + 255) / 256, 256, 0, stream>>>(out);
}